// VectorQuantizer_21019569946793
// MI455X (gfx1250) — compile-verified
//
#include <hip/hip_runtime.h>
#include <hip/hip_bf16.h>
#include <math.h>

// ---------------------------------------------------------------------------
// VQ-VAE vector quantizer for MI455X (gfx1250, wave32).
//   z_e:      [32, 64, 32, 32] f32   (B, C=D, H, W)  -> N = 32768 vectors, D = 64
//   codebook: [1024, 64] f32         (K, D)
// Outputs (flat): q_out[2097152], loss[1], perplexity[1], encodings[32768*1024]
//
// Distance GEMM on V_WMMA_F32_16X16X4_F32 with codes-as-rows so the argmin is
// lane-local. Codebook tiles are staged into LDS with
// GLOBAL_LOAD_ASYNC_TO_LDS_B128 (double-buffered, shared by the 8 waves of a
// block) -> 8x less global codebook traffic; A-fragments come from ds_load
// with a 68-float padded row stride (bank-conflict-free). A-fragments are
// preloaded into registers ahead of the WMMA chain so the DS loads stay in
// flight instead of draining dscnt before every WMMA.
// ---------------------------------------------------------------------------

typedef __attribute__((ext_vector_type(2))) float v2f;
typedef __attribute__((ext_vector_type(8))) float v8f;

#define VQ_K     1024
#define VQ_D     64
#define VQ_N     32768      // 32 * 32 * 32 latent vectors
#define VQ_HW    1024       // 32*32 spatial
#define VQ_ELEMS 2097152    // B*C*H*W
#define VQ_CT    (VQ_K / 16)   // 64 code tiles
#define SWAP16_PATTERN 0x401F  // group-of-32 swizzle: and=0x1f, or=0, xor=0x10

// --- codebook row norms: norms[k] = sum_d cb[k][d]^2 -----------------------
__global__ __launch_bounds__(256) void vq_norms_kernel(
    const float* __restrict__ cb, float* __restrict__ norms) {
  int k = blockIdx.x * blockDim.x + threadIdx.x;
  if (k < VQ_K) {
    const float* row = cb + k * VQ_D;
    float s = 0.0f;
#pragma unroll 8
    for (int d = 0; d < VQ_D; ++d) s += row[d] * row[d];
    norms[k] = s;
  }
}

// --- WMMA distance + argmin ------------------------------------------------
// One wave per tile of 16 z-vectors; 8 waves (one block) share LDS-staged
// codebook tiles. D-matrix = codebook_tile x z_tile: rows = 16 codes,
// cols = 16 z-vectors. Lane l holds column n = l&15, half h = l>>4; VGPR r of
// the accumulator is code  ct*16 + 8*h + r.
__global__ __launch_bounds__(256) void vq_argmin_kernel(
    const float* __restrict__ z_e, const float* __restrict__ cb,
    const float* __restrict__ cb_norm, int* __restrict__ out_idx) {
  constexpr int ROWP = 68;  // padded row stride in floats (bank-conflict-free)
  __shared__ float acb[2][16 * ROWP];

  const int tid  = threadIdx.x;
  const int lane = tid & 31;
  const int wave = (blockIdx.x * blockDim.x + tid) >> 5;  // z-tile id
  const int half = lane >> 4;
  const int n    = lane & 15;

  // ---- async staging assignment: each of 256 threads moves one 16B chunk --
  const int srow = tid >> 4;         // code row within tile (0..15)
  const int scol = (tid & 15) << 2;  // float offset within row (0,4,..,60)
  const float* stage_src0 = cb + (size_t)srow * VQ_D + scol;
  const unsigned lds_dst0 =
      (unsigned)(size_t)&acb[0][srow * ROWP + scol];  // low 32b = LDS offset
  const unsigned lds_dst1 = (unsigned)(size_t)&acb[1][srow * ROWP + scol];

  // prefetch code tile 0
  asm volatile("global_load_async_to_lds_b128 %0, %1, off"
               :: "v"(lds_dst0), "v"(stage_src0) : "memory");

  // ---- z fragments: z_flat[nglob][c] at z_e[b*65536 + c*1024 + sp] --------
  const int b  = wave >> 6;          // tile / (HW/16)
  const int sp = (wave & 63) << 4;   // spatial base of this tile
  const float* zp = z_e + (size_t)b * (VQ_D * VQ_HW) + sp + n;

  // B fragment for k-step s:  .x : K = 4s+2h   .y : K = 4s+2h+1  (column n)
  v2f bfrag[16];
#pragma unroll
  for (int s = 0; s < 16; ++s) {
    const int c0 = 4 * s + 2 * half;
    v2f f;
    f.x = zp[(size_t)c0 * VQ_HW];
    f.y = zp[(size_t)(c0 + 1) * VQ_HW];
    bfrag[s] = f;
  }

  float best_val = 3.402823466e+38f;
  int   best_idx = 0;

  for (int ct = 0; ct < VQ_CT; ++ct) {
    // tile ct landed (own async loads) ...
    asm volatile("s_wait_asynccnt 0x0" ::: "memory");
    // ... and is visible to all waves; also: every wave finished reading
    // buffer (ct+1)&1 during iteration ct-1, so it is free to refill.
    __syncthreads();

    if (ct + 1 < VQ_CT) {
      const float* src = stage_src0 + (size_t)(ct + 1) * (16 * VQ_D);
      const unsigned dst = ((ct + 1) & 1) ? lds_dst1 : lds_dst0;
      asm volatile("global_load_async_to_lds_b128 %0, %1, off"
                   :: "v"(dst), "v"(src) : "memory");
    }

    // A fragments from LDS: row (code) = n ; floats 4s+2h, 4s+2h+1.
    // Load all 16 first (independent ds_loads stay in flight), then run the
    // dependent WMMA accumulation chain.
    const float* ab = &acb[ct & 1][n * ROWP + 2 * half];
    v2f afrag[16];
#pragma unroll
    for (int s = 0; s < 16; ++s) {
      afrag[s] = *(const v2f*)(ab + 4 * s);  // ds_load_b64, conflict-free
    }

    v8f acc = {0.f, 0.f, 0.f, 0.f, 0.f, 0.f, 0.f, 0.f};
#pragma unroll
    for (int s = 0; s < 16; ++s) {
      acc = __builtin_amdgcn_wmma_f32_16x16x4_f32(
          /*neg_a=*/false, afrag[s], /*neg_b=*/false, bfrag[s],
          /*c_mod=*/(short)0, acc, /*reuse_a=*/false, /*reuse_b=*/false);
    }

    const int cbase = ct * 16 + 8 * half;
#pragma unroll
    for (int r = 0; r < 8; ++r) {
      const int   code = cbase + r;
      const float val  = cb_norm[code] - 2.0f * acc[r];  // ||e||^2 - 2 z.e
      const bool  take = (val < best_val) | ((val == best_val) & (code < best_idx));
      best_val = take ? val : best_val;
      best_idx = take ? code : best_idx;
    }
  }

  // Merge the two halves (codes 8h..8h+7 of each tile live in lane and lane^16).
  const int of = __builtin_amdgcn_ds_swizzle(__float_as_int(best_val), SWAP16_PATTERN);
  const int oi = __builtin_amdgcn_ds_swizzle(best_idx, SWAP16_PATTERN);
  const float oval = __int_as_float(of);
  const bool take = (oval < best_val) | ((oval == best_val) & (oi < best_idx));
  best_val = take ? oval : best_val;
  best_idx = take ? oi   : best_idx;

  if (half == 0) out_idx[wave * 16 + n] = best_idx;
}

// --- gather quantized vectors (transposed out) + commitment-loss partial ---
__global__ __launch_bounds__(256) void vq_gather_loss_kernel(
    const float* __restrict__ z_e, const float* __restrict__ cb,
    const int* __restrict__ idx, float* __restrict__ q_out,
    float* __restrict__ loss_acc) {
  __shared__ float red[256];
  const int t  = blockIdx.x * 256 + threadIdx.x;   // flat [B,C,H,W] index
  const int b  = t >> 16;
  const int c  = (t >> 10) & 63;
  const int sp = t & 1023;
  const int n  = (b << 10) | sp;
  const int k  = idx[n];
  const float q = cb[k * VQ_D + c];
  const float z = z_e[t];
  q_out[t] = q;                     // straight-through forward value
  const float d = q - z;
  red[threadIdx.x] = d * d;
  __syncthreads();
#pragma unroll
  for (int s = 128; s > 0; s >>= 1) {
    if (threadIdx.x < s) red[threadIdx.x] += red[threadIdx.x + s];
    __syncthreads();
  }
  if (threadIdx.x == 0) atomicAdd(loss_acc, red[0]);
}

// --- histogram + one-hot scatter (encodings pre-zeroed by memset) ----------
__global__ __launch_bounds__(256) void vq_hist_scatter_kernel(
    const int* __restrict__ idx, int* __restrict__ counts,
    float* __restrict__ enc) {
  const int n = blockIdx.x * blockDim.x + threadIdx.x;  // 0..N-1
  const int k = idx[n];
  atomicAdd(&counts[k], 1);
  enc[(size_t)n * VQ_K + k] = 1.0f;
}

// --- finalize: perplexity + loss scaling -----------------------------------
__global__ __launch_bounds__(1024) void vq_finalize_kernel(
    const int* __restrict__ counts, const float* __restrict__ loss_acc,
    float* __restrict__ loss_out, float* __restrict__ perp_out) {
  __shared__ float red[1024];
  const int k = threadIdx.x;
  const float p = (float)counts[k] * (1.0f / (float)VQ_N);
  red[k] = p * logf(p + 1e-10f);
  __syncthreads();
#pragma unroll
  for (int s = 512; s > 0; s >>= 1) {
    if (k < s) red[k] += red[k + s];
    __syncthreads();
  }
  if (k == 0) {
    *perp_out = expf(-red[0]);
    *loss_out = 0.25f * (*loss_acc) * (1.0f / (float)VQ_ELEMS);
  }
}

extern "C" void kernel_launch(void* const* d_in, const int* in_sizes, int n_in,
                              void* d_out, int out_size, void* d_ws, size_t ws_size,
                              hipStream_t stream) {
  (void)in_sizes; (void)n_in; (void)out_size; (void)ws_size;
  const float* z_e = (const float*)d_in[0];
  const float* cb  = (const float*)d_in[1];

  float* out      = (float*)d_out;
  float* q_out    = out;                    // 2,097,152 floats
  float* loss_out = out + VQ_ELEMS;         // 1
  float* perp_out = out + VQ_ELEMS + 1;     // 1
  float* enc      = out + VQ_ELEMS + 2;     // 32768*1024 floats

  char*  ws        = (char*)d_ws;
  int*   ws_idx    = (int*)ws;                                  // 32768 ints
  float* ws_norm   = (float*)(ws + (size_t)VQ_N * 4);           // 1024 floats
  int*   ws_counts = (int*)(ws + (size_t)VQ_N * 4 + VQ_K * 4);  // 1024 ints
  float* ws_loss   = (float*)(ws + (size_t)VQ_N * 4 + VQ_K * 8);// 1 float

  // Zero counts + loss accumulator (contiguous), and the 128MB one-hot region.
  hipMemsetAsync(ws_counts, 0, (size_t)VQ_K * 4 + 4, stream);
  hipMemsetAsync(enc, 0, (size_t)VQ_N * VQ_K * sizeof(float), stream);

  vq_norms_kernel<<<VQ_K / 256, 256, 0, stream>>>(cb, ws_norm);
  // 2048 z-tiles, 8 per block -> 256 blocks
  vq_argmin_kernel<<<256, 256, 0, stream>>>(z_e, cb, ws_norm, ws_idx);
  vq_gather_loss_kernel<<<VQ_ELEMS / 256, 256, 0, stream>>>(z_e, cb, ws_idx,
                                                            q_out, ws_loss);
  vq_hist_scatter_kernel<<<VQ_N / 256, 256, 0, stream>>>(ws_idx, ws_counts, enc);
  vq_finalize_kernel<<<1, 1024, 0, stream>>>(ws_counts, ws_loss, loss_out, perp_out);
}